// AffinityLoss_70875550318911
// MI455X (gfx1250) — compile-verified
//
#include <hip/hip_runtime.h>

typedef __attribute__((ext_vector_type(2))) float v2f;
typedef __attribute__((ext_vector_type(8))) float v8f;

#define C_  19
#define H_  384
#define W_  384
#define HP_ 382
#define WP_ 382
#define NB_ 2
#define HW_ (H_ * W_)
#define IGNORE_IDX (-100)

#define TJ_  64            // output locations per block (one row segment)
#define NTJ  ((WP_ + TJ_ - 1) / TJ_)          // 6 column tiles
#define LW_  68            // padded tile width (need 66 cols; 68 => di-step 16 mod 64)
#define CP_  20            // padded channel count (19 -> 20, slot 19 zeroed)
#define NBLK (NB_ * HP_ * NTJ)                // 4584 blocks

// Block = 256 threads = 8 waves; one (n, i, j-tile) per block.
// Stage logits tile (19ch x 3rows x 66cols) channel-innermost into LDS with
// coalesced global loads, then each wave computes 8 locations: the 9x9 Gram
// P^T P via 5 chained V_WMMA_F32_16X16X4_F32 (K 19->20 zero-padded), operands
// from unconditional, conflict-free ds_load_b64 (padding lanes read an LDS
// zero region, so EXEC stays untouched), then BCE-with-logits straight from
// the WMMA accumulator layout with LDS label lookups.
__global__ __launch_bounds__(256) void affinity_loss_kernel(
    const float* __restrict__ logits,
    const int*   __restrict__ labels,
    float*       __restrict__ partials)
{
    __shared__ float slog[3 * LW_ * CP_];   // [(di*LW_+col)*CP_ + c], 16.3 KB
    __shared__ float szero[CP_];            // zero operands for padding lanes
    __shared__ int   slab[3 * LW_ + 1];     // labels tile + IGNORE sentinel
    __shared__ float ssum[8];

    const int tid  = threadIdx.x;
    const int lane = tid & 31;
    const int half = lane >> 4;             // K-chunk half of the wave
    const int a    = lane & 15;             // patch position (A row / B col)
    const bool a_ok = (a < 9);
    const int di_a = a_ok ? (a / 3) : 0;
    const int dj_a = a_ok ? (a % 3) : 0;

    // block -> (n, i, j0)
    const int bid = blockIdx.x;
    const int n   = bid / (HP_ * NTJ);
    const int r   = bid % (HP_ * NTJ);
    const int i   = r / NTJ;
    const int j0  = (r % NTJ) * TJ_;
    const int nbase = n * C_ * HW_;

    // ---- stage logits tile: coalesced along columns ----
    for (int e = tid; e < C_ * 3 * 66; e += 256) {
        const int c   = e / 198;            // 198 = 3*66
        const int rem = e % 198;
        const int di  = rem / 66;
        const int col = rem % 66;
        const int gj  = min(j0 + col, W_ - 1);   // clamp (clamped cols never used)
        slog[(di * LW_ + col) * CP_ + c] =
            logits[nbase + c * HW_ + (i + di) * W_ + gj];
    }
    // zero channel-19 padding + stage labels tile
    for (int e = tid; e < 3 * 66; e += 256) {
        const int di  = e / 66;
        const int col = e % 66;
        slog[(di * LW_ + col) * CP_ + (CP_ - 1)] = 0.f;
        const int gj = min(j0 + col, W_ - 1);
        slab[di * LW_ + col] = labels[n * HW_ + (i + di) * W_ + gj];
    }
    if (tid < CP_) szero[tid] = 0.f;
    if (tid == 0)  slab[3 * LW_] = IGNORE_IDX;
    __syncthreads();

    const int w = tid >> 5;                 // wave id in block (0..7)
    float acc = 0.f;

    for (int jl = 0; jl < 8; ++jl) {        // 8 locations per wave
        const int jloc = w * 8 + jl;
        const int j = j0 + jloc;
        if (j >= WP_) continue;             // wave-uniform

        // per-lane operand pointer; padding lanes read LDS zeros -> no EXEC ops
        const float* op = a_ok
            ? &slog[(di_a * LW_ + (jloc + dj_a)) * CP_]
            : &szero[0];

        // issue all 5 ds_load_b64 first so dscnt waits pipeline against WMMA
        v2f ab[5];
#pragma unroll
        for (int kk = 0; kk < 5; ++kk)
            ab[kk] = *(const v2f*)(op + kk * 4 + half * 2);

        // ---- Gram via 5 chained f32 WMMAs; A == B for the symmetric case ----
        v8f cacc = {0.f, 0.f, 0.f, 0.f, 0.f, 0.f, 0.f, 0.f};
#pragma unroll
        for (int kk = 0; kk < 5; ++kk)
            cacc = __builtin_amdgcn_wmma_f32_16x16x4_f32(
                false, ab[kk], false, ab[kk], (short)0, cacc, false, false);

        // ---- BCE over the 81 valid entries; labels straight from LDS ----
        const int b = a;
        const int lb = slab[a_ok ? (di_a * LW_ + (jloc + dj_a)) : (3 * LW_)];
#pragma unroll
        for (int v = 0; v < 8; ++v) {
            const int arow = v + half * 8;
            const bool valid = (arow < 9) && (b < 9);
            const int laoff = valid ? ((arow / 3) * LW_ + (arow % 3)) : 0;
            const int la = slab[laoff + jloc];       // broadcast LDS read
            const float y = (la != IGNORE_IDX && lb != IGNORE_IDX && la == lb)
                                ? 1.f : 0.f;
            const float x = cacc[v];
            const float t = fmaxf(x, 0.f) - x * y
                          + __logf(1.f + __expf(-fabsf(x)));
            acc += valid ? t : 0.f;
        }
    }

    // wave32 + block reduction -> one partial per block
#pragma unroll
    for (int o = 16; o > 0; o >>= 1) acc += __shfl_xor(acc, o, 32);
    if (lane == 0) ssum[w] = acc;
    __syncthreads();
    if (tid == 0) {
        float s = 0.f;
#pragma unroll
        for (int k = 0; k < 8; ++k) s += ssum[k];
        partials[blockIdx.x] = s;
    }
}

__global__ __launch_bounds__(256) void reduce_partials(
    const float* __restrict__ p, int n, float scale, float* __restrict__ out)
{
    __shared__ float s[256];
    float acc = 0.f;
    for (int i = threadIdx.x; i < n; i += 256) acc += p[i];
    s[threadIdx.x] = acc;
    __syncthreads();
    for (int st = 128; st > 0; st >>= 1) {
        if ((int)threadIdx.x < st) s[threadIdx.x] += s[threadIdx.x + st];
        __syncthreads();
    }
    if (threadIdx.x == 0) out[0] = s[0] * scale;
}

extern "C" void kernel_launch(void* const* d_in, const int* in_sizes, int n_in,
                              void* d_out, int out_size, void* d_ws, size_t ws_size,
                              hipStream_t stream)
{
    const float* logits = (const float*)d_in[0];
    const int*   labels = (const int*)d_in[1];
    float* partials = (float*)d_ws;    // NBLK floats ~= 18 KB

    affinity_loss_kernel<<<NBLK, 256, 0, stream>>>(logits, labels, partials);

    const float scale =
        (float)(1.0 / ((double)NB_ * 81.0 * (double)(HP_ * WP_)));
    reduce_partials<<<1, 256, 0, stream>>>(partials, NBLK, scale, (float*)d_out);
}